// ArticleAwareAtten_74302934221488
// MI455X (gfx1250) — compile-verified
//
#include <hip/hip_runtime.h>

typedef __attribute__((ext_vector_type(2))) float v2f;
typedef __attribute__((ext_vector_type(4))) float v4f;
typedef __attribute__((ext_vector_type(8))) float v8f;

#define BB 16
#define SS 2048
#define HH 1024
#define AA 768

// -------- per-wave 16xK * Kx16 f32 GEMM tile via V_WMMA_F32_16X16X4_F32 ----
// A: [16 x K] row-major (lda)
// B: if BT==false: [K x N] row-major (ldb); if BT==true: B[k][n] = Bm[n*ldb+k]
// D: 16 x 16 tile written at columns [n0, n0+16) with row stride ldd
// ISA layout (CDNA5 ISA §7.12.2):
//   A frag (2 VGPR): lanes 0-15 -> M=lane, K = k+0 / k+1 ; lanes 16-31 -> K = k+2 / k+3
//   B frag (2 VGPR): lanes 0-15 -> N=lane, K = k+0 / k+1 ; lanes 16-31 -> K = k+2 / k+3
//   D frag (8 VGPR): VGPR v = row (v + 8*hi), col = lane&15
template <bool BT>
__device__ inline void wmma_tile_f32(const float* __restrict__ Am, int lda,
                                     const float* __restrict__ Bm, int ldb,
                                     float* __restrict__ Dm, int ldd,
                                     int K, int n0) {
  const int lane = threadIdx.x & 31;
  const int lo = lane & 15;
  const int hi = lane >> 4;
  v8f c = {0.f, 0.f, 0.f, 0.f, 0.f, 0.f, 0.f, 0.f};
  for (int k = 0; k < K; k += 4) {
    v2f a;
    a.x = Am[(size_t)lo * lda + k + 2 * hi];
    a.y = Am[(size_t)lo * lda + k + 2 * hi + 1];
    v2f b;
    if (BT) {
      const float* bp = Bm + (size_t)(n0 + lo) * ldb + k + 2 * hi;
      b.x = bp[0];
      b.y = bp[1];
    } else {
      b.x = Bm[(size_t)(k + 2 * hi) * ldb + n0 + lo];
      b.y = Bm[(size_t)(k + 2 * hi + 1) * ldb + n0 + lo];
    }
    // 8 args: (neg_a, A, neg_b, B, c_mod, C, reuse_a, reuse_b)
    c = __builtin_amdgcn_wmma_f32_16x16x4_f32(false, a, false, b, (short)0, c,
                                              false, false);
  }
#pragma unroll
  for (int v = 0; v < 8; ++v) {
    Dm[(size_t)(v + 8 * hi) * ldd + n0 + lo] = c[v];
  }
}

// artT[b,h] = sum_a art[b,a] * transArt[a,h]        (16x768 @ 768x1024)
__global__ void k_gemm_artT(const float* __restrict__ art,
                            const float* __restrict__ transArt,
                            float* __restrict__ artT) {
  wmma_tile_f32<false>(art, AA, transArt, HH, artT, HH, AA, blockIdx.x * 16);
}

// w[b,h] = sum_k transM[h,k] * artT[b,k]  ==  artT @ transM^T
__global__ void k_gemm_w(const float* __restrict__ artT,
                         const float* __restrict__ transM,
                         float* __restrict__ w) {
  wmma_tile_f32<true>(artT, HH, transM, HH, w, HH, HH, blockIdx.x * 16);
}

// out[b, 0:H] = sum_h u[b,h] * transM[h, :]   (u @ transM), row stride 2H
__global__ void k_gemm_out(const float* __restrict__ u,
                           const float* __restrict__ transM,
                           float* __restrict__ out) {
  wmma_tile_f32<false>(u, HH, transM, HH, out, 2 * HH, HH, blockIdx.x * 16);
}

// out[b, H:2H] = artT[b,:] ; zero u accumulator
__global__ void k_prep(const float* __restrict__ artT, float* __restrict__ out,
                       float* __restrict__ u) {
  int i = blockIdx.x * blockDim.x + threadIdx.x;  // 0..16383
  int b = i >> 10;
  int h = i & (HH - 1);
  out[(size_t)b * 2 * HH + HH + h] = artT[i];
  u[i] = 0.f;
}

// logits[b,s] = encs[b,s,:] . w[b,:]   — streaming pass #1 over encs (128 MB)
// one wave per sequence row, float4 coalesced loads, shuffle reduction
__global__ void k_logits(const float* __restrict__ encs,
                         const float* __restrict__ w,
                         float* __restrict__ logits) {
  const int wave = threadIdx.x >> 5;
  const int lane = threadIdx.x & 31;
  const int r = blockIdx.x * 8 + wave;  // 0 .. B*S-1
  const int b = r >> 11;                // r / S
  const float* er = encs + (size_t)r * HH;
  const float* wr = w + (size_t)b * HH;
  float acc = 0.f;
#pragma unroll
  for (int it = 0; it < 8; ++it) {
    int idx = it * 128 + lane * 4;
    v4f e = *(const v4f*)(er + idx);
    v4f wv = *(const v4f*)(wr + idx);
    acc += e.x * wv.x + e.y * wv.y + e.z * wv.z + e.w * wv.w;
  }
#pragma unroll
  for (int off = 16; off > 0; off >>= 1) acc += __shfl_down(acc, off, 32);
  if (lane == 0) logits[r] = acc;
}

// softmax over S per batch row; in-place logits -> p
__global__ void k_softmax(float* __restrict__ logits) {
  __shared__ float red[256];
  const int b = blockIdx.x;
  const int t = threadIdx.x;
  float* row = logits + (size_t)b * SS;
  float vals[8];
  float m = -3.402823466e38f;
#pragma unroll
  for (int j = 0; j < 8; ++j) {
    vals[j] = row[t + j * 256];
    m = fmaxf(m, vals[j]);
  }
  red[t] = m;
  __syncthreads();
  for (int o = 128; o > 0; o >>= 1) {
    if (t < o) red[t] = fmaxf(red[t], red[t + o]);
    __syncthreads();
  }
  m = red[0];
  __syncthreads();
  float sum = 0.f;
#pragma unroll
  for (int j = 0; j < 8; ++j) {
    vals[j] = __expf(vals[j] - m);
    sum += vals[j];
  }
  red[t] = sum;
  __syncthreads();
  for (int o = 128; o > 0; o >>= 1) {
    if (t < o) red[t] += red[t + o];
    __syncthreads();
  }
  float inv = 1.f / red[0];
#pragma unroll
  for (int j = 0; j < 8; ++j) row[t + j * 256] = vals[j] * inv;
}

// u[b,h] += sum_s p[b,s] * encs[b,s,h]  — streaming pass #2 over encs
// (encs should still be L2-resident: 128 MB < 192 MB)
// grid: (b=16, hchunk=4, sblock=8), 256 threads, f32 atomic accumulate
__global__ void k_accum_u(const float* __restrict__ encs,
                          const float* __restrict__ p, float* __restrict__ u) {
  const int b = blockIdx.x;
  const int hc = blockIdx.y;
  const int sb = blockIdx.z;
  const int h = hc * 256 + threadIdx.x;
  const float* pb = p + (size_t)b * SS + sb * 256;
  const float* eb = encs + ((size_t)b * SS + (size_t)sb * 256) * HH + h;
  float acc = 0.f;
  for (int i = 0; i < 256; ++i) {
    acc += pb[i] * eb[(size_t)i * HH];
  }
  atomicAdd(&u[(size_t)b * HH + h], acc);
}

extern "C" void kernel_launch(void* const* d_in, const int* in_sizes, int n_in,
                              void* d_out, int out_size, void* d_ws,
                              size_t ws_size, hipStream_t stream) {
  const float* encs = (const float*)d_in[0];      // [16, 2048, 1024]
  const float* art_vecs = (const float*)d_in[1];  // [16, 768]
  const float* transM = (const float*)d_in[2];    // [1024, 1024]
  const float* transArt = (const float*)d_in[3];  // [768, 1024]
  float* out = (float*)d_out;                     // [16, 2048]

  char* ws = (char*)d_ws;
  float* artT = (float*)(ws);                  // 16*1024 f32 = 64 KB
  float* w = (float*)(ws + (64 << 10));        // 16*1024 f32 = 64 KB
  float* p = (float*)(ws + (128 << 10));       // 16*2048 f32 = 128 KB
  float* u = (float*)(ws + (256 << 10));       // 16*1024 f32 = 64 KB

  // 1) artT = art_vecs @ transArt            (WMMA f32, 64 waves)
  k_gemm_artT<<<64, 32, 0, stream>>>(art_vecs, transArt, artT);
  // 2) w = artT @ transM^T                   (WMMA f32, 64 waves)
  k_gemm_w<<<64, 32, 0, stream>>>(artT, transM, w);
  // 3) out[:, H:2H] = artT ; u = 0
  k_prep<<<64, 256, 0, stream>>>(artT, out, u);
  // 4) logits = encs . w   (bandwidth pass #1, 128 MB)
  k_logits<<<(BB * SS) / 8, 256, 0, stream>>>(encs, w, p);
  // 5) p = softmax(logits) over S
  k_softmax<<<BB, 256, 0, stream>>>(p);
  // 6) u = sum_s p * encs  (bandwidth pass #2, L2-resident)
  k_accum_u<<<dim3(BB, 4, 8), 256, 0, stream>>>(encs, p, u);
  // 7) out[:, 0:H] = u @ transM              (WMMA f32, 64 waves)
  k_gemm_out<<<64, 32, 0, stream>>>(u, transM, out);
}